// NodeNetwork_26439818674552
// MI455X (gfx1250) — compile-verified
//
#include <hip/hip_runtime.h>
#include <math.h>

// ---------------------------------------------------------------------------
// NodeNetwork for MI455X (gfx1250, wave32).
//   Phase 1: edge scatter with global f32 atomics (L2-resident accumulators).
//   Phase 2: fused 4-layer MLP using V_WMMA_F32_16X16X4_F32 (exact fp32) and
//            hardware V_TANH_F32 when available.
// Block = 256 threads (8 waves) = 2 M-tiles x 4 N-tiles over 32 nodes.
// Workspace layout: mi[N*128] | mo[N*128]  (needs 51.2 MB of d_ws).
// ---------------------------------------------------------------------------

typedef float v2f __attribute__((ext_vector_type(2)));
typedef float v8f __attribute__((ext_vector_type(8)));

#define DFEAT 128
#define KCAT  384           // 3*DFEAT
#define LDH0  388           // 384 + 4  -> stride % 64 == 4, conflict-free A reads
#define LDH   132           // 128 + 4
#define MROWS 32            // nodes per block

__device__ __forceinline__ float fast_tanh(float v) {
#if __has_builtin(__builtin_amdgcn_tanhf)
  return __builtin_amdgcn_tanhf(v);          // V_TANH_F32 (gfx1250 TRANS op)
#elif __has_builtin(__builtin_amdgcn_tanh_f32)
  return __builtin_amdgcn_tanh_f32(v);
#else
  return tanhf(v);
#endif
}

// -------------------------------- zero ws ----------------------------------
__global__ void nn_zero_kernel(float* __restrict__ p, long long n) {
  long long i = (long long)blockIdx.x * blockDim.x + threadIdx.x;
  long long stride = (long long)gridDim.x * blockDim.x;
  for (; i < n; i += stride) p[i] = 0.0f;
}

// ------------------------------ edge scatter -------------------------------
// One wave32 per edge; lane handles 4 consecutive features (float4 gather,
// 4 scalar f32 atomic adds per direction -> global_atomic_add_f32).
__global__ void nn_edge_scatter(const float* __restrict__ x,
                                const float* __restrict__ e,
                                const int*   __restrict__ ei,   // [2,E]
                                float* __restrict__ mi,
                                float* __restrict__ mo,
                                int E) {
  int gid  = blockIdx.x * blockDim.x + threadIdx.x;
  int j    = gid >> 5;
  int lane = gid & 31;
  if (j >= E) return;

  int   src = ei[j];
  int   dst = ei[E + j];
  float ev  = e[j];
  int   c   = lane << 2;

  const float4 xs = *(const float4*)(x + (long long)src * DFEAT + c);
  const float4 xd = *(const float4*)(x + (long long)dst * DFEAT + c);

  float* mip = mi + (long long)dst * DFEAT + c;
  float* mop = mo + (long long)src * DFEAT + c;

  atomicAdd(mip + 0, ev * xs.x);
  atomicAdd(mip + 1, ev * xs.y);
  atomicAdd(mip + 2, ev * xs.z);
  atomicAdd(mip + 3, ev * xs.w);
  atomicAdd(mop + 0, ev * xd.x);
  atomicAdd(mop + 1, ev * xd.y);
  atomicAdd(mop + 2, ev * xd.z);
  atomicAdd(mop + 3, ev * xd.w);
}

// ---------------- K=128 dual-tile GEMM (one wave, 16x32 output) ------------
// A-frag lane layout (ISA 7.12.2): lane L, reg r -> A[M=L&15][K=k0+2*(L>>4)+r]
// B-frag analogous on K rows; C/D: lane L, reg r -> [M=r+8*(L>>4)][N=L&15].
// One shared A fragment feeds two WMMAs (adjacent 16-wide N tiles).
__device__ __forceinline__ void nn_gemm128(const float* Hin,            // LDS, stride LDH
                                           const float* __restrict__ W, // [128,128] k-major
                                           int mt, int lane, int n0a,
                                           v8f& acc0, v8f& acc1) {
  const int mrow  = (mt << 4) + (lane & 15);
  const int ncol  = lane & 15;
  const int khalf = (lane >> 4) << 1;
  v8f z = {};
  acc0 = z;
  acc1 = z;
#pragma unroll 4
  for (int k0 = 0; k0 < DFEAT; k0 += 4) {
    const int k = k0 + khalf;
    v2f a, b0, b1;
    a[0]  = Hin[mrow * LDH + k];
    a[1]  = Hin[mrow * LDH + k + 1];
    b0[0] = W[(k    ) * DFEAT + n0a + ncol];
    b0[1] = W[(k + 1) * DFEAT + n0a + ncol];
    b1[0] = W[(k    ) * DFEAT + n0a + 16 + ncol];
    b1[1] = W[(k + 1) * DFEAT + n0a + 16 + ncol];
    acc0 = __builtin_amdgcn_wmma_f32_16x16x4_f32(false, a, false, b0,
                                                 (short)0, acc0, false, false);
    acc1 = __builtin_amdgcn_wmma_f32_16x16x4_f32(false, a, false, b1,
                                                 (short)0, acc1, false, false);
  }
}

// Bias + hardware tanh + store both 16x16 tiles to LDS (stride LDH).
__device__ __forceinline__ void nn_store_lds(float* Hout, int mt, int lane, int n0a,
                                             const v8f& acc0, const v8f& acc1,
                                             float bias0, float bias1) {
  const int ncol = lane & 15;
#pragma unroll
  for (int r = 0; r < 8; ++r) {
    const int m = (mt << 4) + r + ((lane >> 4) << 3);
    Hout[m * LDH + n0a + ncol]      = fast_tanh(acc0[r] + bias0);
    Hout[m * LDH + n0a + 16 + ncol] = fast_tanh(acc1[r] + bias1);
  }
}

// ------------------------------- fused MLP ---------------------------------
__global__ __launch_bounds__(256) void nn_mlp_kernel(
    const float* __restrict__ x,
    const float* __restrict__ mi,
    const float* __restrict__ mo,
    const float* __restrict__ W1, const float* __restrict__ B1,
    const float* __restrict__ W2, const float* __restrict__ B2,
    const float* __restrict__ W3, const float* __restrict__ B3,
    const float* __restrict__ W4, const float* __restrict__ B4,
    float* __restrict__ out, int N) {
  // Single LDS pool: H0 (32x388, layer-1 input) aliases the layer ping-pong
  // buffers Ha/Hb (each 32x132) -- H0 is dead once layer-1 K-loop finishes.
  __shared__ float smem[MROWS * LDH0];          // 12416 floats = 48.5 KB
  float* H0 = smem;
  float* Ha = smem;                             // alias (protected by barrier)
  float* Hb = smem + MROWS * LDH;

  const int tid  = threadIdx.x;
  const int wave = tid >> 5;
  const int lane = tid & 31;
  const int mt   = wave >> 2;                   // 0..1 : M tile
  const int n0a  = (wave & 3) << 5;             // 0,32,64,96 : N tile base
  const int m0   = blockIdx.x * MROWS;
  const int ncol = lane & 15;

  // ---- stage H0 = [mi | mo | x] for 32 nodes (float4, padded rows) ----
  for (int idx = tid; idx < MROWS * (KCAT / 4); idx += 256) {
    const int m  = idx / (KCAT / 4);
    const int c4 = (idx - m * (KCAT / 4)) << 2;
    const long long node = (long long)(m0 + m);
    float4 v = make_float4(0.f, 0.f, 0.f, 0.f);
    if (m0 + m < N) {
      if      (c4 < 128) v = *(const float4*)(mi + node * DFEAT + c4);
      else if (c4 < 256) v = *(const float4*)(mo + node * DFEAT + (c4 - 128));
      else               v = *(const float4*)(x  + node * DFEAT + (c4 - 256));
    }
    *(float4*)(H0 + m * LDH0 + c4) = v;         // rows 16B-aligned (388*4=1552)
  }
  __syncthreads();

  v8f acc0, acc1;

  // ---- layer 1: K = 384 (H0 -> Ha, aliased) ----
  {
    const int mrow  = (mt << 4) + (lane & 15);
    const int khalf = (lane >> 4) << 1;
    v8f z = {};
    acc0 = z;
    acc1 = z;
#pragma unroll 4
    for (int k0 = 0; k0 < KCAT; k0 += 4) {
      const int k = k0 + khalf;
      v2f a, b0, b1;
      a[0]  = H0[mrow * LDH0 + k];
      a[1]  = H0[mrow * LDH0 + k + 1];
      b0[0] = W1[(k    ) * DFEAT + n0a + ncol];
      b0[1] = W1[(k + 1) * DFEAT + n0a + ncol];
      b1[0] = W1[(k    ) * DFEAT + n0a + 16 + ncol];
      b1[1] = W1[(k + 1) * DFEAT + n0a + 16 + ncol];
      acc0 = __builtin_amdgcn_wmma_f32_16x16x4_f32(false, a, false, b0,
                                                   (short)0, acc0, false, false);
      acc1 = __builtin_amdgcn_wmma_f32_16x16x4_f32(false, a, false, b1,
                                                   (short)0, acc1, false, false);
    }
  }
  __syncthreads();   // all H0 reads complete before aliased Ha writes
  nn_store_lds(Ha, mt, lane, n0a, acc0, acc1,
               B1[n0a + ncol], B1[n0a + 16 + ncol]);
  __syncthreads();

  // ---- layer 2: Ha -> Hb ----
  nn_gemm128(Ha, W2, mt, lane, n0a, acc0, acc1);
  nn_store_lds(Hb, mt, lane, n0a, acc0, acc1,
               B2[n0a + ncol], B2[n0a + 16 + ncol]);
  __syncthreads();

  // ---- layer 3: Hb -> Ha ----
  nn_gemm128(Hb, W3, mt, lane, n0a, acc0, acc1);
  nn_store_lds(Ha, mt, lane, n0a, acc0, acc1,
               B3[n0a + ncol], B3[n0a + 16 + ncol]);
  __syncthreads();

  // ---- layer 4: Ha -> global out ----
  nn_gemm128(Ha, W4, mt, lane, n0a, acc0, acc1);
  {
    const float bias0 = B4[n0a + ncol];
    const float bias1 = B4[n0a + 16 + ncol];
#pragma unroll
    for (int r = 0; r < 8; ++r) {
      const int m = (mt << 4) + r + ((lane >> 4) << 3);
      const long long row = (long long)(m0 + m);
      if (m0 + m < N) {
        out[row * DFEAT + n0a + ncol]      = fast_tanh(acc0[r] + bias0);
        out[row * DFEAT + n0a + 16 + ncol] = fast_tanh(acc1[r] + bias1);
      }
    }
  }
}

// ------------------------------- launcher ----------------------------------
extern "C" void kernel_launch(void* const* d_in, const int* in_sizes, int n_in,
                              void* d_out, int out_size, void* d_ws, size_t ws_size,
                              hipStream_t stream) {
  const float* x  = (const float*)d_in[0];
  const float* e  = (const float*)d_in[1];
  const int*   ei = (const int*)  d_in[2];
  const float* W1 = (const float*)d_in[3];
  const float* B1 = (const float*)d_in[4];
  const float* W2 = (const float*)d_in[5];
  const float* B2 = (const float*)d_in[6];
  const float* W3 = (const float*)d_in[7];
  const float* B3 = (const float*)d_in[8];
  const float* W4 = (const float*)d_in[9];
  const float* B4 = (const float*)d_in[10];
  float* out = (float*)d_out;

  const int N = in_sizes[0] / DFEAT;  // 50000
  const int E = in_sizes[1];          // 800000

  float* mi = (float*)d_ws;
  float* mo = mi + (size_t)N * DFEAT;

  // 1) zero accumulators (2 * N * 128 floats)
  const long long ztot = 2LL * N * DFEAT;
  nn_zero_kernel<<<2048, 256, 0, stream>>>(mi, ztot);

  // 2) edge scatter: one wave32 per edge
  const long long ethreads = (long long)E * 32;
  const int eblocks = (int)((ethreads + 255) / 256);
  nn_edge_scatter<<<eblocks, 256, 0, stream>>>(x, e, ei, mi, mo, E);

  // 3) fused MLP: one block per 32 nodes
  const int mblocks = (N + MROWS - 1) / MROWS;  // 1563
  nn_mlp_kernel<<<mblocks, 256, 0, stream>>>(x, mi, mo,
                                             W1, B1, W2, B2, W3, B3, W4, B4,
                                             out, N);
}